// ItemCodeLayer_7765300871693
// MI455X (gfx1250) — compile-verified
//
#include <hip/hip_runtime.h>

// Problem constants (from reference)
#define BATCH      8
#define SEQ        200
#define BS         (BATCH * SEQ)      // 1600 (b,s) rows
#define NUM_ITEMS  100000
#define BYTES      8                  // I
#define VALS       256                // N
#define SUB        64                 // E (K dim)

// Tiling
#define G              8                      // (b,s) rows per block -> 8 x 8KB = 64KB LDS
#define THREADS        256                    // 8 waves (wave32)
#define U              32                     // items per thread
#define ITEMS_PER_TILE (THREADS * U)          // 8192
#define TABLE_STRIDE   (BYTES * VALS)         // 2048 floats per row-table

typedef __attribute__((ext_vector_type(2))) float v2f;
typedef __attribute__((ext_vector_type(8))) float v8f;

__global__ __launch_bounds__(THREADS)
void itemcode_fused_kernel(const float* __restrict__ centroids,   // [8][256][64] f32
                           const int*   __restrict__ input_ids,   // [1600]
                           const int*   __restrict__ item_codes,  // [100002][8]
                           float*       __restrict__ out)         // [1600][100000]
{
    __shared__ float tables[G * TABLE_STRIDE];   // 64 KB: tables[g][i][n]

    const int bsBase   = blockIdx.y * G;
    const int tileBase = blockIdx.x * ITEMS_PER_TILE;
    const int tid   = threadIdx.x;
    const int lane  = tid & 31;
    const int w     = tid >> 5;        // wave id 0..7
    const int ml    = lane & 15;       // M / N lane index within fragment
    const int khalf = lane >> 4;       // 0: K=0,1   1: K=2,3

    // ---------------- Phase 1: build score tables with V_WMMA_F32_16X16X4_F32 -------
    // S[g][i][n] = sum_k centroids[i][code(g,i)][k] * centroids[i][n][k]
    // M-tile = 16 lanes but only 8 real rows (rows 8..15 duplicate 0..7, not stored).
    const int rl = ml & 7;                         // row-in-block (duplicated for ml>=8)
    const int id = input_ids[bsBase + rl];

    for (int jj = 0; jj < 16; ++jj) {
        const int job = jj * 8 + w;                // 128 jobs: i in [0,8) x nTile in [0,16)
        const int i   = job >> 4;
        const int n0  = (job & 15) << 4;
        const int code = item_codes[id * BYTES + i];

        const float* Abase = centroids + i * (VALS * SUB) + code * SUB + khalf * 2;
        const float* Bbase = centroids + i * (VALS * SUB) + (n0 + ml) * SUB + khalf * 2;

        v8f acc = {0.f, 0.f, 0.f, 0.f, 0.f, 0.f, 0.f, 0.f};
        #pragma unroll
        for (int kk = 0; kk < 16; ++kk) {          // K = 64 in steps of 4
            v2f a = *(const v2f*)(Abase + kk * 4); // A[M=ml, K=kk*4 + khalf*2 .. +1]
            v2f b = *(const v2f*)(Bbase + kk * 4); // B[K, N=n0+ml] = C[n0+ml, K]
            acc = __builtin_amdgcn_wmma_f32_16x16x4_f32(
                      /*neg_a=*/false, a, /*neg_b=*/false, b,
                      /*c_mod=*/(short)0, acc, /*reuse_a=*/false, /*reuse_b=*/false);
        }

        // D layout: lanes 0-15 hold M=r in VGPR r (g = r); lanes 16-31 hold M=8+r (dups, drop)
        if (lane < 16) {
            #pragma unroll
            for (int r = 0; r < 8; ++r) {
                tables[r * TABLE_STRIDE + i * VALS + n0 + ml] = acc[r];
            }
        }
    }

    __syncthreads();

    // ---------------- Phase 2: per-item gather-sum, coalesced NT stores -------------
    for (int u = 0; u < U; ++u) {
        const int t = tileBase + u * THREADS + tid;
        if (t < NUM_ITEMS) {
            const int4 ca = *(const int4*)(item_codes + t * BYTES);      // codes 0..3
            const int4 cb = *(const int4*)(item_codes + t * BYTES + 4);  // codes 4..7
            #pragma unroll
            for (int g = 0; g < G; ++g) {
                const float* tb = tables + g * TABLE_STRIDE;
                float v = tb[           ca.x] + tb[1 * VALS + ca.y]
                        + tb[2 * VALS + ca.z] + tb[3 * VALS + ca.w]
                        + tb[4 * VALS + cb.x] + tb[5 * VALS + cb.y]
                        + tb[6 * VALS + cb.z] + tb[7 * VALS + cb.w];
                // 640MB output >> 192MB L2: stream past L2 with a non-temporal store
                __builtin_nontemporal_store(v, out + (size_t)(bsBase + g) * NUM_ITEMS + t);
            }
        }
    }
}

extern "C" void kernel_launch(void* const* d_in, const int* in_sizes, int n_in,
                              void* d_out, int out_size, void* d_ws, size_t ws_size,
                              hipStream_t stream) {
    (void)in_sizes; (void)n_in; (void)out_size; (void)d_ws; (void)ws_size;
    const float* centroids  = (const float*)d_in[0];
    const int*   input_ids  = (const int*)d_in[1];
    const int*   item_codes = (const int*)d_in[2];
    float*       out        = (float*)d_out;

    dim3 grid((NUM_ITEMS + ITEMS_PER_TILE - 1) / ITEMS_PER_TILE,  // 13 item tiles
              BS / G);                                            // 200 row groups
    itemcode_fused_kernel<<<grid, dim3(THREADS), 0, stream>>>(
        centroids, input_ids, item_codes, out);
}